// TypeEnhancedGCNDeepseek_10539849744795
// MI455X (gfx1250) — compile-verified
//
#include <hip/hip_runtime.h>
#include <math.h>

typedef __attribute__((ext_vector_type(2))) float v2f;
typedef __attribute__((ext_vector_type(8))) float v8f;

#define Bq   16
#define Sq   128
#define Dq   300
#define Oq   64
#define Eq   16
#define Tq   45
#define WST  (Dq + Eq)   // 316, row stride of W1/W2

// ---- workspace layout (float offsets) ----
#define HW_OFF  0                       // 2048*300 = 614400
#define E1_OFF  (HW_OFF + Bq*Sq*Dq)     // 45*64 = 2880
#define E2_OFF  (E1_OFF + Tq*Oq)
#define C3_OFF  (E2_OFF + Tq*Oq)        // 45 (pad to 64)
#define HI_OFF  (C3_OFF + 64)           // 2048*64
#define HJ_OFF  (HI_OFF + Bq*Sq*Oq)
#define HT_OFF  (HJ_OFF + Bq*Sq*Oq)
#define AM_OFF  (HT_OFF + Bq*Sq*Oq)     // 16*128*128

// ---------------- kernel 1: position weights + weighted h ----------------
__global__ void pos_hw_kernel(const float* __restrict__ h,
                              const unsigned char* __restrict__ mask,
                              float* __restrict__ hw) {
    __shared__ int s_start, s_end;
    __shared__ float wsh[Sq];
    int b = blockIdx.x, t = threadIdx.x;
    if (t == 0) { s_start = Sq; s_end = -1; }
    __syncthreads();
    if (mask[b * Sq + t]) { atomicMin(&s_start, t); atomicMax(&s_end, t); }
    __syncthreads();
    int st = s_start, en = s_end;
    float invS = 1.0f / (float)Sq;
    float w = (t < st) ? (1.0f - (float)(st - t) * invS)
            : (t > en) ? (1.0f - (float)(t - en) * invS) : 0.0f;
    wsh[t] = w;
    __syncthreads();
    const float* hb = h + (size_t)b * Sq * Dq;
    float* ob = hw + (size_t)b * Sq * Dq;
    for (int idx = t; idx < Sq * Dq; idx += blockDim.x) {
        int s = idx / Dq;
        ob[idx] = hb[idx] * wsh[s];
    }
}

// ---------------- kernel 2: type tables E1, E2, c3 ----------------
__global__ void table_kernel(const float* __restrict__ dep_table,
                             const float* __restrict__ W1,
                             const float* __restrict__ W2,
                             float* __restrict__ E1, float* __restrict__ E2,
                             float* __restrict__ c3) {
    int tid = threadIdx.x;
    for (int i = tid; i < Tq * Oq; i += blockDim.x) {
        int t = i >> 6, o = i & 63;
        float a1 = 0.f, a2 = 0.f;
        #pragma unroll
        for (int e = 0; e < Eq; ++e) {
            float de = dep_table[t * Eq + e];
            a1 += de * W1[o * WST + Dq + e];
            a2 += de * W2[o * WST + Dq + e];
        }
        E1[i] = a1; E2[i] = a2;
    }
    __syncthreads();
    for (int t = tid; t < Tq; t += blockDim.x) {
        float s = 0.f;
        for (int o = 0; o < Oq; ++o) s += E1[t * Oq + o] * E2[t * Oq + o];
        c3[t] = s;
    }
}

// ---------------- kernel 3: projection GEMMs (hi/hj/ht) via f32 WMMA ----------------
// one wave per (m_tile, matrix); acc covers N=0..63 in 4 tiles; K = 300 in steps of 4
__global__ __launch_bounds__(32) void proj_wmma_kernel(
        const float* __restrict__ hw,
        const float* __restrict__ W1, const float* __restrict__ b1,
        const float* __restrict__ W2, const float* __restrict__ b2,
        const float* __restrict__ Wg, const float* __restrict__ bg,
        float* __restrict__ hi, float* __restrict__ hj, float* __restrict__ ht) {
    int mt  = blockIdx.x;   // 0..127
    int mat = blockIdx.y;   // 0..2
    const float* W; const float* bv; float* out; int wstr;
    if (mat == 0)      { W = W1; bv = b1; out = hi; wstr = WST; }
    else if (mat == 1) { W = W2; bv = b2; out = hj; wstr = WST; }
    else               { W = Wg; bv = bg; out = ht; wstr = Dq;  }

    int lane = threadIdx.x;
    int rw   = lane & 15;          // A: row M / B: col N within tile
    int kb   = (lane >> 4) * 2;    // upper half-wave holds K+2,K+3
    int m0   = mt * 16;

    v8f acc[4] = {v8f{}, v8f{}, v8f{}, v8f{}};
    const float* arow = hw + (size_t)(m0 + rw) * Dq;

    for (int k0 = 0; k0 < Dq; k0 += 4) {
        v2f a;
        a.x = arow[k0 + kb];
        a.y = arow[k0 + kb + 1];
        #pragma unroll
        for (int nt = 0; nt < 4; ++nt) {
            int n = nt * 16 + rw;
            v2f bf;
            bf.x = W[n * wstr + k0 + kb];       // B[k][n] = W[n][k]
            bf.y = W[n * wstr + k0 + kb + 1];
            acc[nt] = __builtin_amdgcn_wmma_f32_16x16x4_f32(
                false, a, false, bf, (short)0, acc[nt], false, false);
        }
    }
    // C/D layout: VGPR r -> row m0+r (+8 for lanes 16..31), col = lane&15
    #pragma unroll
    for (int nt = 0; nt < 4; ++nt) {
        #pragma unroll
        for (int r = 0; r < 8; ++r) {
            int mrow = m0 + r + 8 * (lane >> 4);
            int ncol = nt * 16 + rw;
            out[(size_t)mrow * Oq + ncol] = acc[nt][r] + bv[ncol];
        }
    }
}

// ---------------- kernel 4: scores via type-table + masked softmax ----------------
__global__ void score_softmax_kernel(const float* __restrict__ hi,
                                     const float* __restrict__ hj,
                                     const float* __restrict__ E1,
                                     const float* __restrict__ E2,
                                     const float* __restrict__ c3,
                                     const int* __restrict__ dep,
                                     float* __restrict__ Amat) {
    __shared__ float shi[Oq], shj[Oq], tab[Tq], red[Sq];
    __shared__ float c0sh;
    int bs = blockIdx.x, tid = threadIdx.x;
    if (tid < Oq) shi[tid] = hi[(size_t)bs * Oq + tid];
    else          shj[tid - Oq] = hj[(size_t)bs * Oq + (tid - Oq)];
    __syncthreads();
    if (tid < Oq) red[tid] = shi[tid] * shj[tid];
    __syncthreads();
    if (tid == 0) {
        float s = 0.f;
        for (int i = 0; i < Oq; ++i) s += red[i];
        c0sh = s;
    }
    __syncthreads();
    if (tid < Tq) {
        float s = 0.f;
        for (int o = 0; o < Oq; ++o)
            s += shi[o] * E2[tid * Oq + o] + shj[o] * E1[tid * Oq + o];
        tab[tid] = (c0sh + s + c3[tid]) * 0.125f;   // /sqrt(64)
    }
    __syncthreads();
    int typ  = dep[(size_t)bs * Sq + tid];
    float sc = tab[typ];
    // row max over ALL raw scores (reference computes max before masking)
    red[tid] = sc;
    __syncthreads();
    for (int off = 64; off > 0; off >>= 1) {
        if (tid < off) red[tid] = fmaxf(red[tid], red[tid + off]);
        __syncthreads();
    }
    float mx = red[0];
    __syncthreads();
    float ex = (typ != 0) ? __expf(sc - mx) : 0.0f;
    red[tid] = ex;
    __syncthreads();
    for (int off = 64; off > 0; off >>= 1) {
        if (tid < off) red[tid] += red[tid + off];
        __syncthreads();
    }
    float denom = red[0] + 1e-6f;
    Amat[(size_t)bs * Sq + tid] = ex / denom;
}

// ---------------- kernel 5: out = relu(A @ h_trans + bias) via f32 WMMA ----------------
__global__ __launch_bounds__(32) void attn_wmma_kernel(
        const float* __restrict__ Amat, const float* __restrict__ ht,
        const float* __restrict__ bias, float* __restrict__ out) {
    int b  = blockIdx.x;   // 0..15
    int mt = blockIdx.y;   // 0..7
    int lane = threadIdx.x;
    int rw = lane & 15;
    int kb = (lane >> 4) * 2;
    int m0 = mt * 16;
    const float* Ab = Amat + (size_t)b * Sq * Sq;
    const float* hb = ht   + (size_t)b * Sq * Oq;

    v8f acc[4] = {v8f{}, v8f{}, v8f{}, v8f{}};
    for (int k0 = 0; k0 < Sq; k0 += 4) {
        v2f a;
        a.x = Ab[(size_t)(m0 + rw) * Sq + k0 + kb];
        a.y = Ab[(size_t)(m0 + rw) * Sq + k0 + kb + 1];
        #pragma unroll
        for (int nt = 0; nt < 4; ++nt) {
            int n = nt * 16 + rw;
            v2f bf;
            bf.x = hb[(size_t)(k0 + kb) * Oq + n];
            bf.y = hb[(size_t)(k0 + kb + 1) * Oq + n];
            acc[nt] = __builtin_amdgcn_wmma_f32_16x16x4_f32(
                false, a, false, bf, (short)0, acc[nt], false, false);
        }
    }
    float* ob = out + (size_t)b * Sq * Oq;
    #pragma unroll
    for (int nt = 0; nt < 4; ++nt) {
        #pragma unroll
        for (int r = 0; r < 8; ++r) {
            int mrow = m0 + r + 8 * (lane >> 4);
            int ncol = nt * 16 + rw;
            float v = acc[nt][r] + bias[ncol];
            ob[(size_t)mrow * Oq + ncol] = fmaxf(v, 0.0f);
        }
    }
}

extern "C" void kernel_launch(void* const* d_in, const int* in_sizes, int n_in,
                              void* d_out, int out_size, void* d_ws, size_t ws_size,
                              hipStream_t stream) {
    (void)in_sizes; (void)n_in; (void)out_size; (void)ws_size;
    const float* h         = (const float*)d_in[0];
    const float* dep_table = (const float*)d_in[1];
    const float* W1        = (const float*)d_in[2];
    const float* b1        = (const float*)d_in[3];
    const float* W2        = (const float*)d_in[4];
    const float* b2        = (const float*)d_in[5];
    const float* Wg        = (const float*)d_in[6];
    const float* bg        = (const float*)d_in[7];
    const float* bias      = (const float*)d_in[8];
    const int*   dep       = (const int*)d_in[9];
    const unsigned char* amask = (const unsigned char*)d_in[10];

    float* ws  = (float*)d_ws;
    float* hw  = ws + HW_OFF;
    float* E1  = ws + E1_OFF;
    float* E2  = ws + E2_OFF;
    float* c3  = ws + C3_OFF;
    float* hi  = ws + HI_OFF;
    float* hj  = ws + HJ_OFF;
    float* ht  = ws + HT_OFF;
    float* Amt = ws + AM_OFF;
    float* out = (float*)d_out;

    pos_hw_kernel<<<Bq, Sq, 0, stream>>>(h, amask, hw);
    table_kernel<<<1, 256, 0, stream>>>(dep_table, W1, W2, E1, E2, c3);
    proj_wmma_kernel<<<dim3(Bq * Sq / 16, 3), 32, 0, stream>>>(hw, W1, b1, W2, b2,
                                                               Wg, bg, hi, hj, ht);
    score_softmax_kernel<<<Bq * Sq, Sq, 0, stream>>>(hi, hj, E1, E2, c3, dep, Amt);
    attn_wmma_kernel<<<dim3(Bq, Sq / 16), 32, 0, stream>>>(Amt, ht, bias, out);
}